// InfoNCELoss_25726854103480
// MI455X (gfx1250) — compile-verified
//
#include <hip/hip_runtime.h>
#include <math.h>

typedef float v2f __attribute__((ext_vector_type(2)));
typedef float v8f __attribute__((ext_vector_type(8)));
typedef int   v4i __attribute__((vector_size(16)));   // builtin's pointee type

#define B_TOTAL 8192
#define D_DIM   512
#define INV_T   10.0f      // 1 / temperature (0.1)
#define BM      32         // rows per workgroup
#define BN      128        // columns per chunk
#define BK      32         // K slice held in LDS (double buffered)
#define NSLICES (D_DIM / BK)

#if defined(__HIP_DEVICE_COMPILE__) && \
    __has_builtin(__builtin_amdgcn_global_load_async_to_lds_b128) && \
    __has_builtin(__builtin_amdgcn_s_wait_asynccnt)
#define USE_ASYNC 1
#else
#define USE_ASYNC 0
#endif

// Fused InfoNCE: tiled fp32 WMMA GEMM (sim = A * P^T / T) with online
// logsumexp per row; sim tile lives only in LDS, never in HBM.
// K-slices are double buffered and staged with async global->LDS DMA
// (ASYNCcnt path) so loads for slice s+1 overlap WMMA on slice s.
__global__ __launch_bounds__(256) void infonce_fused_wmma(
    const float* __restrict__ anchor,
    const float* __restrict__ positive,
    float* __restrict__ out)
{
    __shared__ float lA[2][BM][BK + 4];    // anchor slices   [buf][m][k]
    __shared__ float lB[2][BN][BK + 4];    // positive slices [buf][n][k]
    __shared__ float lS[BM][BN + 4];       // scaled sim tile
    __shared__ float pmax[BM][8];
    __shared__ float psum[BM][8];
    __shared__ float pdiag[BM];
    __shared__ float plos[BM];

    const int tid  = threadIdx.x;
    const int lane = tid & 31;
    const int wave = tid >> 5;
    const int ms   = blockIdx.x * BM;

    // WMMA fragment coordinates (32-bit A 16x4 layout; B mirrors it)
    const int fm = lane & 15;              // M (A) / N (B) within 16
    const int fk = (lane >> 4) << 1;       // K pair base: 0 or 2

    // 2x8 grid of 16x16 tiles; wave w -> row (w>>2), cols (w&3)*2, +1
    const int arow  = (wave >> 2) * 16 + fm;
    const int bcol0 = ((wave & 3) * 2) * 16 + fm;
    const int bcol1 = bcol0 + 16;

    // online-LSE ownership: 8 threads per row, 16 columns each
    const int rrow = tid >> 3;             // 0..31
    const int oq   = tid & 7;              // 0..7 (column octant)
    float m_run = -INFINITY;
    float s_run = 0.0f;

    // ---- staging helper: one K-slice (A: 256 x b128, B: 1024 x b128) ----
    auto stage = [&](int buf, int ns_, int ks_) {
        {   // A slice: 32 rows x 8 float4 -> one b128 per thread
            int r = tid >> 3, c4 = tid & 7;
            const float* g = anchor + (size_t)(ms + r) * D_DIM + ks_ + c4 * 4;
            float* l = &lA[buf][r][c4 * 4];
#if USE_ASYNC
            __builtin_amdgcn_global_load_async_to_lds_b128((v4i*)g, (v4i*)l, 0, 0);
#else
            float4 v = *(const float4*)g;
            l[0] = v.x; l[1] = v.y; l[2] = v.z; l[3] = v.w;
#endif
        }
        #pragma unroll
        for (int i = 0; i < 4; ++i) {      // B slice: 128 rows x 8 float4
            int idx = tid + i * 256;
            int r = idx >> 3, c4 = idx & 7;
            const float* g = positive + (size_t)(ns_ + r) * D_DIM + ks_ + c4 * 4;
            float* l = &lB[buf][r][c4 * 4];
#if USE_ASYNC
            __builtin_amdgcn_global_load_async_to_lds_b128((v4i*)g, (v4i*)l, 0, 0);
#else
            float4 v = *(const float4*)g;
            l[0] = v.x; l[1] = v.y; l[2] = v.z; l[3] = v.w;
#endif
        }
    };
    auto wait_stage = [&]() {
#if USE_ASYNC
        __builtin_amdgcn_s_wait_asynccnt(0);
#endif
    };

    for (int ns = 0; ns < B_TOTAL; ns += BN) {
        v8f acc0 = {};
        v8f acc1 = {};

        // pipeline prologue: stage slice 0 into buffer 0
        __syncthreads();                   // prior chunk done with both buffers
        stage(0, ns, 0);
        wait_stage();
        __syncthreads();

        int cur = 0;
        #pragma unroll 1
        for (int s = 0; s < NSLICES; ++s) {
            if (s + 1 < NSLICES)
                stage(cur ^ 1, ns, (s + 1) * BK);   // async: overlaps compute

            // ---- WMMA on resident slice: V_WMMA_F32_16X16X4_F32 ----
            #pragma unroll 4
            for (int k = 0; k < BK; k += 4) {
                const int kk = k + fk;
                v2f af = { lA[cur][arow][kk],  lA[cur][arow][kk + 1]  };
                v2f b0 = { lB[cur][bcol0][kk], lB[cur][bcol0][kk + 1] };
                v2f b1 = { lB[cur][bcol1][kk], lB[cur][bcol1][kk + 1] };
                acc0 = __builtin_amdgcn_wmma_f32_16x16x4_f32(
                           false, af, false, b0, (short)0, acc0, false, false);
                acc1 = __builtin_amdgcn_wmma_f32_16x16x4_f32(
                           false, af, false, b1, (short)0, acc1, false, false);
            }

            wait_stage();                  // next buffer's DMA complete (this wave)
            __syncthreads();               // ...and every wave's; also fences reuse
            cur ^= 1;
        }

        // ---- spill scaled sim tile to LDS (C: VGPR v -> M=v(+8), N=lane&15) ----
        {
            const int cmb = (wave >> 2) * 16 + (lane >> 4) * 8;
            const int cn0 = ((wave & 3) * 2) * 16 + (lane & 15);
            #pragma unroll
            for (int v = 0; v < 8; ++v) {
                lS[cmb + v][cn0]      = acc0[v] * INV_T;
                lS[cmb + v][cn0 + 16] = acc1[v] * INV_T;
            }
        }
        __syncthreads();

        // ---- online logsumexp update over this thread's 16 columns ----
        float vbuf[16];
        float mc = -INFINITY;
        #pragma unroll
        for (int j = 0; j < 16; ++j) {
            float v = lS[rrow][oq * 16 + j];
            vbuf[j] = v;
            mc = fmaxf(mc, v);
        }
        float mnew = fmaxf(m_run, mc);
        s_run *= __expf(m_run - mnew);     // exp(-inf)=0 handles first chunk
        #pragma unroll
        for (int j = 0; j < 16; ++j) s_run += __expf(vbuf[j] - mnew);
        m_run = mnew;

        // diagonal sim[i][i] passes through exactly one (chunk, octant)
        const int dcol = (ms + rrow) - ns;
        if (dcol >= oq * 16 && dcol < oq * 16 + 16)
            pdiag[rrow] = lS[rrow][dcol];
        // next chunk's prologue barrier orders these lS reads vs. next spill
    }

    // ---- merge 8 partial LSE states per row, then block reduce ----
    __syncthreads();
    pmax[rrow][oq] = m_run;
    psum[rrow][oq] = s_run;
    __syncthreads();
    if (tid < BM) {
        float M = -INFINITY;
        #pragma unroll
        for (int q = 0; q < 8; ++q) M = fmaxf(M, pmax[tid][q]);
        float S = 0.0f;
        #pragma unroll
        for (int q = 0; q < 8; ++q) S += psum[tid][q] * __expf(pmax[tid][q] - M);
        plos[tid] = (__logf(S) + M) - pdiag[tid];
    }
    __syncthreads();
    if (tid == 0) {
        float a = 0.0f;
        #pragma unroll 4
        for (int i = 0; i < BM; ++i) a += plos[i];
        atomicAdd(out, a * (1.0f / (float)B_TOTAL));   // mean over rows
    }
}

extern "C" void kernel_launch(void* const* d_in, const int* in_sizes, int n_in,
                              void* d_out, int out_size, void* d_ws, size_t ws_size,
                              hipStream_t stream) {
    (void)in_sizes; (void)n_in; (void)d_ws; (void)ws_size; (void)out_size;
    const float* anchor   = (const float*)d_in[0];
    const float* positive = (const float*)d_in[1];
    float* out = (float*)d_out;

    (void)hipMemsetAsync(out, 0, sizeof(float), stream);   // graph-capture legal
    dim3 grid(B_TOTAL / BM);   // 256 workgroups
    infonce_fused_wmma<<<grid, 256, 0, stream>>>(anchor, positive, out);
}